// ContourIntegrationLayer_10136122818690
// MI455X (gfx1250) — compile-verified
//
#include <hip/hip_runtime.h>
#include <stdint.h>

// ---------------------------------------------------------------------------
// Depthwise 5x5 conv (SAME) + gated residual:  out = (conv*alpha + bias)*x + x
// x (32,112,112,96) NHWC f32, kernel (5,5,96), alpha/bias scalars.
// Bandwidth-bound (AI ~6.3 F/B, HBM floor ~13us @23.3TB/s). Data movement:
//   - interior tiles: TDM tensor_load_to_lds (2D strided DMA, TENSORcnt)
//   - border tiles:   per-chunk async global->LDS B128 (ASYNCcnt) + zero halo
// Compute: per-channel register sliding window, each LDS element read once.
// WMMA inapplicable: depthwise has no cross-channel contraction.
// ---------------------------------------------------------------------------

namespace {
constexpr int kB = 32, kH = 112, kW = 112, kC = 96, kN = 5;
constexpr int TILE = 8, HALO = 2, TDIM = TILE + 2 * HALO;   // 12x12 haloed tile
constexpr int THREADS = kC;                                  // 96 = 3 x wave32
constexpr int CHUNKS = TDIM * TDIM * (kC / 4);               // 3456 x 16B
constexpr int ROW_ELEMS = TDIM * kC;                         // 1152 f32 / tile row
constexpr int ROW_BYTES = ROW_ELEMS * 4;                     // 4608 B
}

typedef __attribute__((ext_vector_type(4))) float floatx4;
typedef int v4i __attribute__((vector_size(16)));            // matches builtin
typedef __attribute__((address_space(1))) v4i gv4i;          // global int4
typedef __attribute__((address_space(3))) v4i lv4i;          // LDS int4

typedef unsigned int v4u_e __attribute__((ext_vector_type(4)));
typedef int v8i_e __attribute__((ext_vector_type(8)));
typedef int v4i_e __attribute__((ext_vector_type(4)));

#if defined(__gfx1250__) && __has_builtin(__builtin_amdgcn_tensor_load_to_lds)
#define HAVE_TDM 1
#else
#define HAVE_TDM 0
#endif

// --- CDNA5 async global->LDS (ASYNCcnt-tracked, VGPR-bypassing) -------------
__device__ __forceinline__ void async_load_b128_to_lds(const float* gsrc,
                                                       float* ldst) {
#if defined(__gfx1250__) && __has_builtin(__builtin_amdgcn_global_load_async_to_lds_b128)
  gv4i* gp = (gv4i*)(uintptr_t)gsrc;            // global AS(1), 64-bit
  lv4i* lp = (lv4i*)(unsigned)(uintptr_t)ldst;  // LDS AS(3), 32-bit offset
  __builtin_amdgcn_global_load_async_to_lds_b128(gp, lp, /*offset=*/0,
                                                 /*cpol=*/0);
#elif defined(__HIP_DEVICE_COMPILE__)
  unsigned lds_off = (unsigned)(uintptr_t)ldst;
  asm volatile("global_load_async_to_lds_b128 %0, %1, off"
               :: "v"(lds_off), "v"(gsrc)
               : "memory");
#else
  (void)gsrc; (void)ldst;
#endif
}

__device__ __forceinline__ void wait_asynccnt0() {
#if defined(__gfx1250__) && __has_builtin(__builtin_amdgcn_s_wait_asynccnt)
  __builtin_amdgcn_s_wait_asynccnt(0);
#elif defined(__HIP_DEVICE_COMPILE__)
  asm volatile("s_wait_asynccnt 0" ::: "memory");
#endif
}

__device__ __forceinline__ void wait_tensorcnt0() {
#if defined(__gfx1250__) && __has_builtin(__builtin_amdgcn_s_wait_tensorcnt)
  __builtin_amdgcn_s_wait_tensorcnt(0);
#elif defined(__HIP_DEVICE_COMPILE__)
  asm volatile("s_wait_tensorcnt 0" ::: "memory");
#endif
}

// --- TDM: DMA `rows` image rows (1152 f32 each, stride W*C) into LDS --------
// D# packing per CDNA5 ISA, 2D normal mode.
// 6-arg builtin form (clang-23/therock-10.0): groups 2/3 + trailing 256-bit
// operand are >=3D / gather-mode state -> zero for a 2D transfer.
__device__ __forceinline__ void tdm_load_rows(const float* gsrc_row,
                                              unsigned lds_byte_addr,
                                              int rows) {
#if HAVE_TDM
  const uint64_t ga = (uint64_t)(uintptr_t)gsrc_row;
  v4u_e g0;
  g0[0] = 1u;                                   // count=1, user descriptor
  g0[1] = lds_byte_addr;                        // lds_addr [63:32]
  g0[2] = (unsigned)(ga & 0xffffffffu);         // global_addr [95:64]
  g0[3] = (unsigned)((ga >> 32) & 0x1ffffffu)   // global_addr [120:96]
          | 0x80000000u;                        // type=2 ('10') [127:126]
  v8i_e g1;
  g1[0] = (int)(0x2u << 16);                    // wg_mask=0, data_size=4B
  g1[1] = (int)((unsigned)ROW_ELEMS << 16);     // tensor_dim0[15:0]
  g1[2] = (int)((unsigned)rows << 16);          // tdim0[31:16]=0|tensor_dim1.lo
  g1[3] = (int)((unsigned)ROW_ELEMS << 16);     // tdim1.hi=0 | tile_dim0
  g1[4] = rows;                                 // tile_dim1 | tile_dim2=0
  g1[5] = kW * kC;                              // tensor_dim0_stride lo32
  g1[6] = 0;                                    // stride0 hi | stride1 lo
  g1[7] = 0;                                    // stride1 hi
  v4i_e gz4 = {0, 0, 0, 0};                     // groups 2/3: dims unused
  v8i_e gz8 = {0, 0, 0, 0, 0, 0, 0, 0};         // trailing operand: unused
  __builtin_amdgcn_tensor_load_to_lds(g0, g1, gz4, gz4, gz8, /*cpol=*/0);
#else
  (void)gsrc_row; (void)lds_byte_addr; (void)rows;
#endif
}

// ---------------------------------------------------------------------------
__global__ __launch_bounds__(THREADS)
void dwconv5x5_gate_kernel(const float* __restrict__ x,
                           const float* __restrict__ kern,
                           const float* __restrict__ alpha,
                           const float* __restrict__ bias,
                           float* __restrict__ out) {
  __shared__ float tile[TDIM * TDIM * kC];  // 55,296 B

  const int tid = threadIdx.x;              // == channel c
  const int bid = blockIdx.x;
  const int tw = bid % (kW / TILE);
  const int th = (bid / (kW / TILE)) % (kH / TILE);
  const int b  = bid / ((kW / TILE) * (kH / TILE));
  const int h0 = th * TILE, w0 = tw * TILE;

  const bool interior =
      (unsigned)(th - 1) < (unsigned)(kH / TILE - 2) &&
      (unsigned)(tw - 1) < (unsigned)(kW / TILE - 2);

  if (HAVE_TDM && interior) {
    // ---- One 2D TDM descriptor per wave: wave w DMAs tile rows 4w..4w+3 ---
    const int wave = tid >> 5;                              // 0..2
    const int gh = h0 - HALO + 4 * wave;
    const float* gsrc = x + (((size_t)b * kH + gh) * kW + (w0 - HALO)) * kC;
    const unsigned lds_base =
        (unsigned)(uintptr_t)&tile[0] + (unsigned)(wave * 4 * ROW_BYTES);
    tdm_load_rows(gsrc, lds_base, 4);
  } else {
    // ---- Border: per-chunk async B128 with SAME zero padding --------------
    for (int i = tid; i < CHUNKS; i += THREADS) {
      const int q  = i % (kC / 4);          // 16B chunk within pixel
      const int p  = i / (kC / 4);          // tile pixel 0..143
      const int pw = p % TDIM;
      const int ph = p / TDIM;
      const int gh = h0 - HALO + ph;
      const int gw = w0 - HALO + pw;
      float* ldst = &tile[p * kC + q * 4];
      if ((unsigned)gh < (unsigned)kH && (unsigned)gw < (unsigned)kW) {
        const float* gsrc = x + (((size_t)b * kH + gh) * kW + gw) * kC + q * 4;
        async_load_b128_to_lds(gsrc, ldst);
      } else {
        *(floatx4*)ldst = (floatx4)0.f;
      }
    }
  }

  // ---- 25 per-channel weights + scalars into registers (overlaps DMA) -----
  const int c = tid;
  float wk[kN * kN];
#pragma unroll
  for (int t = 0; t < kN * kN; ++t) wk[t] = kern[t * kC + c];
  const float av = alpha[0];
  const float bv = bias[0];

  wait_tensorcnt0();
  wait_asynccnt0();
  __syncthreads();

  // ---- 5-row register sliding window: each LDS element read exactly once --
  float win[kN][TDIM];
#pragma unroll
  for (int r = 0; r < kN - 1; ++r)
#pragma unroll
    for (int j = 0; j < TDIM; ++j)
      win[r][j] = tile[(r * TDIM + j) * kC + c];

#pragma unroll
  for (int oh = 0; oh < TILE; ++oh) {
    // roll in tile row (oh+4)
#pragma unroll
    for (int j = 0; j < TDIM; ++j)
      win[(oh + 4) % kN][j] = tile[((oh + 4) * TDIM + j) * kC + c];

#pragma unroll
    for (int ow = 0; ow < TILE; ++ow) {
      float acc = 0.f;
#pragma unroll
      for (int dy = 0; dy < kN; ++dy)
#pragma unroll
        for (int dx = 0; dx < kN; ++dx)
          acc = fmaf(wk[dy * kN + dx], win[(oh + dy) % kN][ow + dx], acc);
      const float xv   = win[(oh + 2) % kN][ow + 2];   // center = input pixel
      const float gate = fmaf(acc, av, bv);
      out[(((size_t)b * kH + (h0 + oh)) * kW + (w0 + ow)) * kC + c] =
          fmaf(gate, xv, xv);                          // gate*x + x
    }
  }
}

// ---------------------------------------------------------------------------
extern "C" void kernel_launch(void* const* d_in, const int* in_sizes, int n_in,
                              void* d_out, int out_size, void* d_ws,
                              size_t ws_size, hipStream_t stream) {
  const float* x     = (const float*)d_in[0];
  const float* kern  = (const float*)d_in[1];
  const float* alpha = (const float*)d_in[2];
  const float* bias  = (const float*)d_in[3];
  float* out = (float*)d_out;

  dim3 grid(kB * (kH / TILE) * (kW / TILE));   // 32*14*14 = 6272 blocks
  dim3 block(THREADS);                         // 96 threads = 3 wave32
  hipLaunchKernelGGL(dwconv5x5_gate_kernel, grid, block, 0, stream,
                     x, kern, alpha, bias, out);
}